// SE_attn_fusion_26757646254180
// MI455X (gfx1250) — compile-verified
//
#include <hip/hip_runtime.h>

// Problem constants (match the reference)
#define BB   32
#define HN   12
#define N1   256
#define N2   320
#define SS   256
#define ROWS (BB * HN * N1)          // 98304  (b,h,n1) rows
#define BH   (BB * HN)               // 384    (b,h) rows
#define EPS  1e-5f
#define TOT  ((long long)ROWS * N2)  // elements per attn tensor

typedef float v2f __attribute__((ext_vector_type(2)));
typedef float v8f __attribute__((ext_vector_type(8)));

// LDS row strides (dwords). %64 == 4 -> A-fragment column reads (16 lanes,
// consecutive rows, same column) land in distinct banks; multiple of 4 keeps
// 16-byte alignment for float4 fills and 8-byte alignment for v2f reads.
#define LDV 516
#define LDH 260

// ---------------------------------------------------------------------------
// Kernel 1: rowwise max over N2 + permutation scatter into vex[BH][512]
// one wave32 per (b,h,n1) row
// ---------------------------------------------------------------------------
__global__ void k_max_scatter(const float* __restrict__ rgb,
                              const float* __restrict__ tir,
                              const int*   __restrict__ gidx,
                              float*       __restrict__ vex) {
  const int wave = (blockIdx.x * blockDim.x + threadIdx.x) >> 5;
  const int lane = threadIdx.x & 31;
  if (wave >= ROWS) return;
  const long long base = (long long)wave * N2;
  float mr = -3.402823466e38f, mt = -3.402823466e38f;
  #pragma unroll
  for (int j = 0; j < N2 / 32; ++j) {
    const long long e = base + lane + 32 * j;
    mr = fmaxf(mr, rgb[e]);
    mt = fmaxf(mt, tir[e]);
  }
  #pragma unroll
  for (int off = 16; off > 0; off >>= 1) {
    mr = fmaxf(mr, __shfl_xor(mr, off, 32));
    mt = fmaxf(mt, __shfl_xor(mt, off, 32));
  }
  if (lane == 0) {
    const int n1 = wave & (N1 - 1);
    const int bh = wave >> 8;          // N1 == 256
    const int b  = bh / HN;
    const int s  = gidx[b * N1 + n1];  // permutation -> collision free
    vex[bh * (2 * SS) + s]      = mr;
    vex[bh * (2 * SS) + SS + s] = mt;
  }
}

// ---------------------------------------------------------------------------
// Kernel 2: fused LayerNorm(512) + Linear(512,256)+ReLU + Linear(256,512)
//           + sigmoid, via V_WMMA_F32_16X16X4_F32. 128 threads (4 waves),
//           one block per 16-row M tile (24 blocks total).
// ---------------------------------------------------------------------------
__global__ void k_ln_mlp(const float* __restrict__ vex,
                         const float* __restrict__ lng,
                         const float* __restrict__ lnb,
                         const float* __restrict__ W1,
                         const float* __restrict__ b1,
                         const float* __restrict__ W2,
                         const float* __restrict__ b2,
                         float*       __restrict__ gout) {
  __shared__ float sv[16 * LDV];   // normalized inputs  [16][512]
  __shared__ float sh[16 * LDH];   // hidden activations [16][256]

  const int tid     = threadIdx.x;
  const int wave    = tid >> 5;
  const int lane    = tid & 31;
  const int ln      = lane & 15;        // N index / M index inside fragment
  const int halfsel = lane >> 4;        // K-half selector
  const int row0    = blockIdx.x * 16;  // first (b,h) row of this tile

  // ---- load 16 x 512 fp32 tile into LDS (float4, fully coalesced) ----
  {
    const float4* src = (const float4*)(vex + (long long)row0 * 512);
    for (int q = tid; q < 16 * 128; q += 128) {
      const int r  = q >> 7;
      const int c4 = (q & 127) << 2;
      *(float4*)(&sv[r * LDV + c4]) = src[q];
    }
  }
  __syncthreads();

  // ---- LayerNorm: wave w owns rows 4w..4w+3 ----
  #pragma unroll
  for (int rr = 0; rr < 4; ++rr) {
    const int r = wave * 4 + rr;
    float s = 0.f, sq = 0.f;
    #pragma unroll
    for (int j = 0; j < 16; ++j) {
      const float x = sv[r * LDV + lane + 32 * j];
      s += x; sq += x * x;
    }
    #pragma unroll
    for (int off = 16; off > 0; off >>= 1) {
      s  += __shfl_xor(s,  off, 32);
      sq += __shfl_xor(sq, off, 32);
    }
    const float mu   = s * (1.0f / 512.0f);
    const float var  = sq * (1.0f / 512.0f) - mu * mu;
    const float rinv = rsqrtf(var + EPS);
    #pragma unroll
    for (int j = 0; j < 16; ++j) {
      const int c = lane + 32 * j;
      sv[r * LDV + c] = (sv[r * LDV + c] - mu) * rinv * lng[c] + lnb[c];
    }
  }
  __syncthreads();

  // ---- GEMM1: h[16,256] = relu(v[16,512] @ W1[512,256] + b1) ----
  // wave w computes 4 N-tiles: n0 = (w*4+i)*16
  {
    v8f zero = {};
    v8f acc[4];
    #pragma unroll
    for (int i = 0; i < 4; ++i) acc[i] = zero;

    for (int kk = 0; kk < 512 / 4; ++kk) {
      const int kb = kk * 4 + 2 * halfsel;
      v2f a;                                  // A: M=ln, K=kb..kb+1
      a.x = sv[ln * LDV + kb];
      a.y = sv[ln * LDV + kb + 1];
      #pragma unroll
      for (int i = 0; i < 4; ++i) {
        const int n = (wave * 4 + i) * 16 + ln;
        v2f bfr;                              // B: K=kb..kb+1, N=n
        bfr.x = W1[kb * 256 + n];
        bfr.y = W1[(kb + 1) * 256 + n];
        acc[i] = __builtin_amdgcn_wmma_f32_16x16x4_f32(
            false, a, false, bfr, (short)0, acc[i], false, false);
      }
    }
    #pragma unroll
    for (int i = 0; i < 4; ++i) {
      const int n  = (wave * 4 + i) * 16 + ln;
      const float bias = b1[n];
      #pragma unroll
      for (int vg = 0; vg < 8; ++vg) {
        const int m = vg + 8 * halfsel;       // D layout
        sh[m * LDH + n] = fmaxf(acc[i][vg] + bias, 0.0f);
      }
    }
  }
  __syncthreads();

  // ---- GEMM2: g[16,512] = sigmoid(h[16,256] @ W2[256,512] + b2) ----
  // wave w computes 8 N-tiles: n0 = (w*8+i)*16
  {
    v8f zero = {};
    v8f acc[8];
    #pragma unroll
    for (int i = 0; i < 8; ++i) acc[i] = zero;

    for (int kk = 0; kk < 256 / 4; ++kk) {
      const int kb = kk * 4 + 2 * halfsel;
      v2f a;
      a.x = sh[ln * LDH + kb];
      a.y = sh[ln * LDH + kb + 1];
      #pragma unroll
      for (int i = 0; i < 8; ++i) {
        const int n = (wave * 8 + i) * 16 + ln;
        v2f bfr;
        bfr.x = W2[kb * 512 + n];
        bfr.y = W2[(kb + 1) * 512 + n];
        acc[i] = __builtin_amdgcn_wmma_f32_16x16x4_f32(
            false, a, false, bfr, (short)0, acc[i], false, false);
      }
    }
    #pragma unroll
    for (int i = 0; i < 8; ++i) {
      const int n  = (wave * 8 + i) * 16 + ln;
      const float bias = b2[n];
      #pragma unroll
      for (int vg = 0; vg < 8; ++vg) {
        const int m = vg + 8 * halfsel;
        const float x = acc[i][vg] + bias;
        gout[(long long)(row0 + m) * 512 + n] = 1.0f / (1.0f + __expf(-x));
      }
    }
  }
}

// ---------------------------------------------------------------------------
// Kernel 3: gather gates through permutation and modulate both tensors.
// one 320-thread block per (b,h,n1) row. Output order: (tir, rgb).
// ---------------------------------------------------------------------------
__global__ void k_modulate(const float* __restrict__ rgb,
                           const float* __restrict__ tir,
                           const int*   __restrict__ gidx,
                           const float* __restrict__ g,
                           float*       __restrict__ out) {
  const int r  = blockIdx.x;
  const int n1 = r & (N1 - 1);
  const int bh = r >> 8;
  const int b  = bh / HN;
  const int s  = gidx[b * N1 + n1];
  const float grgb = g[bh * 512 + s];
  const float gtir = g[bh * 512 + SS + s];
  const long long e = (long long)r * N2 + threadIdx.x;
  out[e]       = gtir * tir[e];
  out[TOT + e] = grgb * rgb[e];
}

// ---------------------------------------------------------------------------
extern "C" void kernel_launch(void* const* d_in, const int* in_sizes, int n_in,
                              void* d_out, int out_size, void* d_ws, size_t ws_size,
                              hipStream_t stream) {
  (void)in_sizes; (void)n_in; (void)out_size; (void)ws_size;
  const float* attn_rgb = (const float*)d_in[0];
  const float* attn_tir = (const float*)d_in[1];
  const int*   gidx     = (const int*)  d_in[2];
  const float* lng      = (const float*)d_in[3];
  const float* lnb      = (const float*)d_in[4];
  const float* W1       = (const float*)d_in[5];
  const float* b1       = (const float*)d_in[6];
  const float* W2       = (const float*)d_in[7];
  const float* b2       = (const float*)d_in[8];
  float* out = (float*)d_out;

  float* vex = (float*)d_ws;            // [384][512]
  float* g   = vex + BH * 2 * SS;       // [384][512]

  // 98304 rows, one wave each, 8 waves per 256-thread block
  k_max_scatter<<<ROWS / 8, 256, 0, stream>>>(attn_rgb, attn_tir, gidx, vex);
  // 24 M-tiles of 16 rows, 4 waves per block
  k_ln_mlp<<<BH / 16, 128, 0, stream>>>(vex, lng, lnb, W1, b1, W2, b2, g);
  // one block per (b,h,n1) row
  k_modulate<<<ROWS, N2, 0, stream>>>(attn_rgb, attn_tir, gidx, g, out);
}